// SubSamplingLayer_69372311765440
// MI455X (gfx1250) — compile-verified
//
#include <hip/hip_runtime.h>
#include <hip/hip_bf16.h>
#include <math.h>

// ---------------------------------------------------------------------------
// NUFFT-style subsampling layer for MI455X (gfx1250, wave32, WMMA).
//
//   out = M * (scatter(gather(ksp))) * M^T          (complex, per batch)
//   M[n,k] = exp(+2*pi*i * (k-160)*(n-160)/320)     (folds both fftshifts,
//                                                    H*W cancels ifft norm)
//
// GEMMs run on V_WMMA_F32_16X16X4_F32 with 2x2 tile register blocking
// (32x32 complex macro-tile per wave) to halve L2 fragment traffic.
// Scatter uses wave-uniform aggregation to kill atomic hot-spotting from
// degenerate (skipped) trajectory points.
// ---------------------------------------------------------------------------

#define RES     320
#define NB      32
#define PLANE   (RES * RES)          // 102400 floats per real/imag plane
#define NPTS    (RES * RES)          // trajectory points
#define MT1D    (RES / 32)           // 10 macro-tiles (32x32) per dim
#define MT2D    (MT1D * MT1D)        // 100 macro-tiles per image

typedef __attribute__((ext_vector_type(2))) float v2f;
typedef __attribute__((ext_vector_type(8))) float v8f;

// ---------------------------------------------------------------------------
// Kernel 0: zero the scatter grid and build the shifted-IDFT twiddle tables.
// Phase reduced mod 320 in exact integer math before sincosf (raw phase
// would reach ~160*pi and destroy fp32 accuracy).
// ---------------------------------------------------------------------------
__global__ void __launch_bounds__(256)
setup_kernel(float* __restrict__ Mr, float* __restrict__ Mi,
             float* __restrict__ kgrid /* NB*2*PLANE floats */) {
    int idx = blockIdx.x * blockDim.x + threadIdx.x;
    int total = NB * 2 * PLANE;
    if (idx < total) kgrid[idx] = 0.0f;
    if (idx < PLANE) {
        int n = idx / RES;
        int k = idx % RES;
        int prod = (k - RES / 2) * (n - RES / 2);
        int m = prod % RES; if (m < 0) m += RES;
        float ang = (6.2831853071795864769f / (float)RES) * (float)m;
        float s, c;
        sincosf(ang, &s, &c);
        Mr[idx] = c;
        Mi[idx] = s;
    }
}

// ---------------------------------------------------------------------------
// Kernel 1: bilinear gather + scatter-add.
// One thread per trajectory point, batch loop inside (corner data is batch-
// invariant). Degenerate trajectory points (the masked region) all collapse
// onto one grid cell -> naive atomics serialize on one address. The wave-
// uniform fast path (batch-invariant check) reduces 32 lanes' contributions
// with a shfl_xor butterfly and issues a single atomic per corner per wave.
// ---------------------------------------------------------------------------
__global__ void __launch_bounds__(256)
grid_kernel(const float* __restrict__ in,    // (B,1,RES,RES,2)
            const float* __restrict__ traj,  // (NPTS,2)
            float* __restrict__ kgrid) {     // (B,{R,I},PLANE)
    int p = blockIdx.x * blockDim.x + threadIdx.x;
    if (p >= NPTS) return;
    int lane = threadIdx.x & 31;

    float tx = traj[2 * p + 0];
    float ty = traj[2 * p + 1];
    float gx = tx * (2.0f / (float)RES);
    float gy = ty * (2.0f / (float)RES);
    float x = ((gx + 1.0f) * (float)RES - 1.0f) * 0.5f;
    float y = ((gy + 1.0f) * (float)RES - 1.0f) * 0.5f;
    float x0f = floorf(x), y0f = floorf(y);
    float fx = x - x0f,  fy = y - y0f;
    int x0 = (int)x0f,   y0 = (int)y0f;

    const int   cx[4] = { x0, x0 + 1, x0,     x0 + 1 };
    const int   cy[4] = { y0, y0,     y0 + 1, y0 + 1 };
    const float cw[4] = { (1.0f - fx) * (1.0f - fy), fx * (1.0f - fy),
                          (1.0f - fx) * fy,          fx * fy };
    int  cidx[4];
    bool cv[4];
    unsigned vbits = 0;
#pragma unroll
    for (int i = 0; i < 4; ++i) {
        cv[i] = (cx[i] >= 0) && (cx[i] < RES) && (cy[i] >= 0) && (cy[i] < RES);
        vbits |= (cv[i] ? 1u : 0u) << i;
        int ix = min(max(cx[i], 0), RES - 1);
        int iy = min(max(cy[i], 0), RES - 1);
        cidx[i] = iy * RES + ix;
    }

    // Batch-invariant wave-uniformity: same (x0,y0,validity) across all 32
    // lanes => all 4 corner addresses identical wave-wide.
    unsigned key = ((unsigned)(x0 + 1) & 0x3ffu)
                 | (((unsigned)(y0 + 1) & 0x3ffu) << 10)
                 | (vbits << 20);
    bool uniform = __all(key == __shfl(key, 0, 32));

    for (int b = 0; b < NB; ++b) {
        const float* src = in + (size_t)b * 2 * PLANE;  // interleaved (re,im)
        float sr = 0.0f, si = 0.0f;
#pragma unroll
        for (int i = 0; i < 4; ++i) {
            if (cv[i]) {
                v2f v = *(const v2f*)(src + 2 * cidx[i]);  // 8B-aligned
                sr += v.x * cw[i];
                si += v.y * cw[i];
            }
        }
        float* kR = kgrid + (size_t)b * 2 * PLANE;
        float* kI = kR + PLANE;
        if (uniform) {
#pragma unroll
            for (int i = 0; i < 4; ++i) {
                if (cv[i]) {
                    float vr = sr * cw[i];
                    float vi = si * cw[i];
#pragma unroll
                    for (int off = 16; off > 0; off >>= 1) {
                        vr += __shfl_xor(vr, off, 32);
                        vi += __shfl_xor(vi, off, 32);
                    }
                    if (lane == 0) {
                        atomicAdd(&kR[cidx[i]], vr);
                        atomicAdd(&kI[cidx[i]], vi);
                    }
                }
            }
        } else {
#pragma unroll
            for (int i = 0; i < 4; ++i) {
                if (cv[i]) {
                    atomicAdd(&kR[cidx[i]], sr * cw[i]);
                    atomicAdd(&kI[cidx[i]], si * cw[i]);
                }
            }
        }
    }
}

// ---------------------------------------------------------------------------
// WMMA fragment layouts (ISA 7.12.2, wave32):
//   A 16x4 f32 : lane l (0..15) row M=l, VGPR{0,1}=K{0,1}; lanes 16..31 K{2,3}
//   B 4x16 f32 : lane l (0..15) col N=l, VGPR{0,1}=K{0,1}; lanes 16..31 K{2,3}
//   C/D 16x16  : VGPR r, lanes 0..15 row r, lanes 16..31 row r+8; col=lane&15
// Each wave computes a 32x32 complex macro-tile: 2x2 sub-tiles x 4 real
// accumulators (RR,II,RI,IR) = 16 v8f accumulators, 16 WMMA per K-step on
// 8 v2f fragment loads (4 B of L2 traffic per WMMA per lane).
// ---------------------------------------------------------------------------

// Kernel 2: tmp[b] = kgrid[b] * M^T   (tmp[r,x] = sum_k kg[r,k] * M[x,k])
__global__ void __launch_bounds__(256)
pass1_kernel(const float* __restrict__ kgrid, const float* __restrict__ Mr,
             const float* __restrict__ Mi, float* __restrict__ tmp) {
    int wave = (int)((blockIdx.x * blockDim.x + threadIdx.x) >> 5);
    int lane = threadIdx.x & 31;
    int b  = wave / MT2D;
    int t  = wave % MT2D;
    int tm = t / MT1D, tn = t % MT1D;
    int half = lane >> 4;
    int l    = lane & 15;

    const float* aR = kgrid + (size_t)b * 2 * PLANE;
    const float* aI = aR + PLANE;
    const int arow[2] = { tm * 32 + l, tm * 32 + 16 + l };
    const int brow[2] = { tn * 32 + l, tn * 32 + 16 + l };  // B[k][n]=M[n][k]

    v8f acc[2][2][4];   // [mi][ni][{RR,II,RI,IR}]
#pragma unroll
    for (int mi = 0; mi < 2; ++mi)
#pragma unroll
        for (int ni = 0; ni < 2; ++ni)
#pragma unroll
            for (int c = 0; c < 4; ++c) acc[mi][ni][c] = (v8f){};

    for (int k0 = 0; k0 < RES; k0 += 4) {
        int ka = k0 + 2 * half;               // 8B-aligned element offset
        v2f ar[2], ai[2], br[2], bi[2];
#pragma unroll
        for (int mi = 0; mi < 2; ++mi) {
            ar[mi] = *(const v2f*)(aR + (size_t)arow[mi] * RES + ka);
            ai[mi] = *(const v2f*)(aI + (size_t)arow[mi] * RES + ka);
        }
#pragma unroll
        for (int ni = 0; ni < 2; ++ni) {
            br[ni] = *(const v2f*)(Mr + (size_t)brow[ni] * RES + ka);
            bi[ni] = *(const v2f*)(Mi + (size_t)brow[ni] * RES + ka);
        }
#pragma unroll
        for (int mi = 0; mi < 2; ++mi)
#pragma unroll
            for (int ni = 0; ni < 2; ++ni) {
                acc[mi][ni][0] = __builtin_amdgcn_wmma_f32_16x16x4_f32(
                    false, ar[mi], false, br[ni], (short)0, acc[mi][ni][0], false, false);
                acc[mi][ni][1] = __builtin_amdgcn_wmma_f32_16x16x4_f32(
                    false, ai[mi], false, bi[ni], (short)0, acc[mi][ni][1], false, false);
                acc[mi][ni][2] = __builtin_amdgcn_wmma_f32_16x16x4_f32(
                    false, ar[mi], false, bi[ni], (short)0, acc[mi][ni][2], false, false);
                acc[mi][ni][3] = __builtin_amdgcn_wmma_f32_16x16x4_f32(
                    false, ai[mi], false, br[ni], (short)0, acc[mi][ni][3], false, false);
            }
    }

    float* tR = tmp + (size_t)b * 2 * PLANE;
    float* tI = tR + PLANE;
#pragma unroll
    for (int mi = 0; mi < 2; ++mi)
#pragma unroll
        for (int ni = 0; ni < 2; ++ni) {
            int col = tn * 32 + ni * 16 + l;
#pragma unroll
            for (int vr = 0; vr < 8; ++vr) {
                int row = tm * 32 + mi * 16 + vr + 8 * half;
                tR[(size_t)row * RES + col] = acc[mi][ni][0][vr] - acc[mi][ni][1][vr];
                tI[(size_t)row * RES + col] = acc[mi][ni][2][vr] + acc[mi][ni][3][vr];
            }
        }
}

// Kernel 3: out[b] = M * tmp[b]   (out[y,x] = sum_r M[y,r] * tmp[r,x])
// Stores straight into the reference (B,1,2,RES,RES) output layout.
__global__ void __launch_bounds__(256)
pass2_kernel(const float* __restrict__ tmp, const float* __restrict__ Mr,
             const float* __restrict__ Mi, float* __restrict__ out) {
    int wave = (int)((blockIdx.x * blockDim.x + threadIdx.x) >> 5);
    int lane = threadIdx.x & 31;
    int b  = wave / MT2D;
    int t  = wave % MT2D;
    int tm = t / MT1D, tn = t % MT1D;
    int half = lane >> 4;
    int l    = lane & 15;

    const int arow[2] = { tm * 32 + l, tm * 32 + 16 + l };       // A=M rows
    const int bcol[2] = { tn * 32 + l, tn * 32 + 16 + l };       // B=tmp cols
    const float* tR = tmp + (size_t)b * 2 * PLANE;
    const float* tI = tR + PLANE;

    v8f acc[2][2][4];
#pragma unroll
    for (int mi = 0; mi < 2; ++mi)
#pragma unroll
        for (int ni = 0; ni < 2; ++ni)
#pragma unroll
            for (int c = 0; c < 4; ++c) acc[mi][ni][c] = (v8f){};

    for (int k0 = 0; k0 < RES; k0 += 4) {
        int ka = k0 + 2 * half;
        v2f ar[2], ai[2], br[2], bi[2];
#pragma unroll
        for (int mi = 0; mi < 2; ++mi) {
            ar[mi] = *(const v2f*)(Mr + (size_t)arow[mi] * RES + ka);
            ai[mi] = *(const v2f*)(Mi + (size_t)arow[mi] * RES + ka);
        }
#pragma unroll
        for (int ni = 0; ni < 2; ++ni) {
            br[ni] = (v2f){ tR[(size_t)(ka + 0) * RES + bcol[ni]],
                            tR[(size_t)(ka + 1) * RES + bcol[ni]] };
            bi[ni] = (v2f){ tI[(size_t)(ka + 0) * RES + bcol[ni]],
                            tI[(size_t)(ka + 1) * RES + bcol[ni]] };
        }
#pragma unroll
        for (int mi = 0; mi < 2; ++mi)
#pragma unroll
            for (int ni = 0; ni < 2; ++ni) {
                acc[mi][ni][0] = __builtin_amdgcn_wmma_f32_16x16x4_f32(
                    false, ar[mi], false, br[ni], (short)0, acc[mi][ni][0], false, false);
                acc[mi][ni][1] = __builtin_amdgcn_wmma_f32_16x16x4_f32(
                    false, ai[mi], false, bi[ni], (short)0, acc[mi][ni][1], false, false);
                acc[mi][ni][2] = __builtin_amdgcn_wmma_f32_16x16x4_f32(
                    false, ar[mi], false, bi[ni], (short)0, acc[mi][ni][2], false, false);
                acc[mi][ni][3] = __builtin_amdgcn_wmma_f32_16x16x4_f32(
                    false, ai[mi], false, br[ni], (short)0, acc[mi][ni][3], false, false);
            }
    }

    float* oR = out + (size_t)b * 2 * PLANE;   // (B,1,2,H,W): ch0=real
    float* oI = oR + PLANE;                    //              ch1=imag
#pragma unroll
    for (int mi = 0; mi < 2; ++mi)
#pragma unroll
        for (int ni = 0; ni < 2; ++ni) {
            int col = tn * 32 + ni * 16 + l;
#pragma unroll
            for (int vr = 0; vr < 8; ++vr) {
                int row = tm * 32 + mi * 16 + vr + 8 * half;
                oR[(size_t)row * RES + col] = acc[mi][ni][0][vr] - acc[mi][ni][1][vr];
                oI[(size_t)row * RES + col] = acc[mi][ni][2][vr] + acc[mi][ni][3][vr];
            }
        }
}

// ---------------------------------------------------------------------------
// Launcher. Workspace layout (floats):
//   [0 , PLANE)                       Mr
//   [PLANE , 2*PLANE)                 Mi
//   [2*PLANE , 2*PLANE + NB*2*PLANE)  kgrid (per batch: R plane, I plane)
//   [... , ... + NB*2*PLANE)          tmp   (same layout)
// Total = (2 + 4*NB)*PLANE*4 bytes ≈ 53.3 MB.
// ---------------------------------------------------------------------------
extern "C" void kernel_launch(void* const* d_in, const int* in_sizes, int n_in,
                              void* d_out, int out_size, void* d_ws, size_t ws_size,
                              hipStream_t stream) {
    const float* ksp  = (const float*)d_in[0];   // (B,1,RES,RES,2) fp32
    const float* traj = (const float*)d_in[1];   // (NPTS,2) fp32

    float* ws    = (float*)d_ws;
    float* Mr    = ws;
    float* Mi    = Mr + PLANE;
    float* kgrid = Mi + PLANE;                        // NB*2*PLANE
    float* tmp   = kgrid + (size_t)NB * 2 * PLANE;    // NB*2*PLANE

    const int zero_total = NB * 2 * PLANE;            // 6 553 600
    setup_kernel<<<(zero_total + 255) / 256, 256, 0, stream>>>(Mr, Mi, kgrid);

    grid_kernel<<<(NPTS + 255) / 256, 256, 0, stream>>>(ksp, traj, kgrid);

    const int waves  = NB * MT2D;                     // 3200 macro-tiles
    const int blocks = waves / 8;                     // 8 waves / 256-thread block
    pass1_kernel<<<blocks, 256, 0, stream>>>(kgrid, Mr, Mi, tmp);
    pass2_kernel<<<blocks, 256, 0, stream>>>(tmp, Mr, Mi, (float*)d_out);
}